// CXLoss_3753801416902
// MI455X (gfx1250) — compile-verified
//
#include <hip/hip_runtime.h>
#include <hip/hip_bf16.h>
#include <math.h>

// Problem constants (from reference): N=2, C=256, H=W=64, P=4096
#define CXN 2
#define CXC 256
#define CXP 4096         // H*W
#define NPOS (CXN * CXP) // 8192 positions total

#define KC  64           // k-chunk staged in LDS
#define LDA 68           // padded LDS row stride (floats): bank-conflict-free frags

typedef __attribute__((ext_vector_type(2))) float v2f;
typedef __attribute__((ext_vector_type(8))) float v8f;

// ds_swizzle_b32, group-of-32 xor mode: offset = (xor_mask<<10) | and_mask(0x1F)
#define SWZ_F32(x, imm) \
    __uint_as_float((unsigned)__builtin_amdgcn_ds_swizzle((int)__float_as_uint(x), (imm)))

__device__ __forceinline__ float grp16_min(float v) { // min within 16-lane groups
    v = fminf(v, SWZ_F32(v, 0x041F)); // xor 1
    v = fminf(v, SWZ_F32(v, 0x081F)); // xor 2
    v = fminf(v, SWZ_F32(v, 0x101F)); // xor 4
    v = fminf(v, SWZ_F32(v, 0x201F)); // xor 8
    return v;
}
__device__ __forceinline__ float grp16_sum(float v) { // sum within 16-lane groups
    v += SWZ_F32(v, 0x041F);
    v += SWZ_F32(v, 0x081F);
    v += SWZ_F32(v, 0x101F);
    v += SWZ_F32(v, 0x201F);
    return v;
}

// ---------------------------------------------------------------------------
// Kernel 1: per-channel mean of featureT over (N,H,W). One block per channel.
// ---------------------------------------------------------------------------
__global__ void cx_mean_kernel(const float* __restrict__ fT, float* __restrict__ meanT) {
    const int c = blockIdx.x;
    const int t = threadIdx.x;
    float s = 0.0f;
    for (int j = t; j < NPOS; j += 256) {
        const int n = j >> 12;
        const int p = j & (CXP - 1);
        s += fT[(size_t)(n * CXC + c) * CXP + p];
    }
    __shared__ float red[256];
    red[t] = s;
    __syncthreads();
    for (int st = 128; st > 0; st >>= 1) {
        if (t < st) red[t] += red[t + st];
        __syncthreads();
    }
    if (t == 0) meanT[c] = red[0] * (1.0f / (float)NPOS);
}

// ---------------------------------------------------------------------------
// Kernel 2: center by meanT, L2-normalize over channels, write [n*P+p][C]
// row-major (C contiguous) so WMMA fragments load as contiguous float2/float4.
// ---------------------------------------------------------------------------
__global__ void cx_normalize_kernel(const float* __restrict__ f,
                                    const float* __restrict__ meanT,
                                    float* __restrict__ out) {
    const int b = blockIdx.x;        // n*4096 + p
    const int n = b >> 12;
    const int p = b & (CXP - 1);
    const int c = threadIdx.x;       // 0..255
    const float v = f[(size_t)(n * CXC + c) * CXP + p] - meanT[c];
    __shared__ float red[256];
    red[c] = v * v;
    __syncthreads();
    for (int st = 128; st > 0; st >>= 1) {
        if (c < st) red[c] += red[c + st];
        __syncthreads();
    }
    __shared__ float rn;
    if (c == 0) rn = rsqrtf(red[0]);
    __syncthreads();
    out[(size_t)b * CXC + c] = v * rn;
}

// ---------------------------------------------------------------------------
// Kernel 3: init per-row/per-col stats (ws is NOT re-initialized by harness).
// ---------------------------------------------------------------------------
__global__ void cx_init_kernel(float* rowmin, float* rowsum, float* colmax) {
    const int i = blockIdx.x * 256 + threadIdx.x;
    if (i < NPOS) {
        rowmin[i] = 3.402823466e+38f; // FLT_MAX
        rowsum[i] = 0.0f;
        colmax[i] = 0.0f;
    }
}

// ---------------------------------------------------------------------------
// LDS-tiled WMMA GEMM passes.
// Block = 512 threads (16 waves) computes a 64x64 tile of
//   dist[n,row,col] = dot(QN[n,row,:], TN[n,col,:]), K=256.
// Per KC=64 chunk: A/B panels (64xKC) staged in LDS via coalesced float4
// loads; each wave computes one 16x16 subtile (wave w -> (w>>2, w&3)) with
// 16x V_WMMA_F32_16X16X4_F32, fragments as bank-conflict-free ds_load_b64.
// A frag (16x4 f32): lane L<16 -> {A[L][k],A[L][k+1]}, lane L+16 -> {+2,+3};
// B (4x16) is transpose-symmetric -> identical addressing on row-major TN.
// C/D: VGPR i -> row i (lanes 0-15) / row i+8 (lanes 16-31), col = lane&15.
// PASS 1: row-wise min of raw=(1-d)/2        -> rowmin (uint-bit atomicMin)
// PASS 2: row-wise sum of Wt=exp((1-rel)*10) -> rowsum (float atomicAdd)
// PASS 3: col-wise max of CX=Wt/rowsum       -> colmax (uint-bit atomicMax)
// ---------------------------------------------------------------------------
template <int PASS>
__global__ __launch_bounds__(512) void cx_gemm_pass(const float* __restrict__ QN,
                                                    const float* __restrict__ TN,
                                                    float* rowmin, float* rowsum,
                                                    float* colmax) {
    __shared__ float ldsA[64 * LDA];
    __shared__ float ldsB[64 * LDA];

    const int tid  = threadIdx.x;
    const int lane = tid & 31;
    const int wave = tid >> 5;     // 0..15
    const int si   = wave >> 2;    // subtile row 0..3
    const int sj   = wave & 3;     // subtile col 0..3
    const int r    = lane & 15;
    const int kh   = lane >> 4;    // K half-select

    const int blk = blockIdx.x;    // 4096 blocks per batch image
    const int n   = blk >> 12;
    const int t   = blk & 4095;
    const int br  = t >> 6;        // 64-row block  (queries)
    const int bc  = t & 63;        // 64-col block  (targets)

    const float* Abase = QN + (size_t)(n * CXP + br * 64) * CXC;
    const float* Bbase = TN + (size_t)(n * CXP + bc * 64) * CXC;

    v8f acc = {};

    for (int k0 = 0; k0 < CXC; k0 += KC) {
        // stage 64xKC panels of A and B (coalesced: 16 float4 per row)
#pragma unroll
        for (int it = 0; it < 2; ++it) {
            const int f4  = tid + it * 512;   // 0..1023
            const int row = f4 >> 4;
            const int c4  = (f4 & 15) * 4;
            const float4 av = *(const float4*)(Abase + (size_t)row * CXC + k0 + c4);
            const float4 bv = *(const float4*)(Bbase + (size_t)row * CXC + k0 + c4);
            *(float4*)(&ldsA[row * LDA + c4]) = av;
            *(float4*)(&ldsB[row * LDA + c4]) = bv;
        }
        __syncthreads();

        const float* aLds = &ldsA[(si * 16 + r) * LDA + kh * 2];
        const float* bLds = &ldsB[(sj * 16 + r) * LDA + kh * 2];
#pragma unroll
        for (int kk = 0; kk < KC; kk += 4) {
            v2f a  = *(const v2f*)(aLds + kk);
            v2f b2 = *(const v2f*)(bLds + kk);
            acc = __builtin_amdgcn_wmma_f32_16x16x4_f32(
                /*neg_a=*/false, a, /*neg_b=*/false, b2,
                /*c_mod=*/(short)0, acc, /*reuse_a=*/false, /*reuse_b=*/false);
        }
        __syncthreads();
    }

    const int rowBase = n * CXP + br * 64 + si * 16;
    const int colBase = n * CXP + bc * 64 + sj * 16;

    if (PASS == 1) {
#pragma unroll
        for (int i = 0; i < 8; ++i) {
            float raw = fmaxf((1.0f - acc[i]) * 0.5f, 0.0f); // >=0 for bit-atomics
            raw = grp16_min(raw);
            if (r == 0)
                atomicMin((unsigned int*)&rowmin[rowBase + i + kh * 8],
                          __float_as_uint(raw));
        }
    } else if (PASS == 2) {
#pragma unroll
        for (int i = 0; i < 8; ++i) {
            const int row = rowBase + i + kh * 8;
            const float m = rowmin[row];
            const float raw = fmaxf((1.0f - acc[i]) * 0.5f, 0.0f);
            const float rel = raw / (m + 1e-5f);
            float wt = __expf((1.0f - rel) * 10.0f); // (B-rel)/SIGMA, B=1, SIGMA=0.1
            wt = grp16_sum(wt);
            if (r == 0) atomicAdd(&rowsum[row], wt);
        }
    } else { // PASS 3
        float cmax = 0.0f;
#pragma unroll
        for (int i = 0; i < 8; ++i) {
            const int row = rowBase + i + kh * 8;
            const float m = rowmin[row];
            const float S = rowsum[row];
            const float raw = fmaxf((1.0f - acc[i]) * 0.5f, 0.0f);
            const float rel = raw / (m + 1e-5f);
            const float wt = __expf((1.0f - rel) * 10.0f);
            cmax = fmaxf(cmax, wt / S);
        }
        // lanes L and L+16 hold the same column (rows 0-7 vs 8-15): combine.
        cmax = fmaxf(cmax, SWZ_F32(cmax, 0x401F)); // xor 16
        if (lane < 16)
            atomicMax((unsigned int*)&colmax[colBase + r], __float_as_uint(cmax));
    }
}

// ---------------------------------------------------------------------------
// Kernel 4: final scalar: mean over n of -log(mean_p colmax[n,p])
// ---------------------------------------------------------------------------
__global__ void cx_final_kernel(const float* __restrict__ colmax, float* __restrict__ out) {
    const int t = threadIdx.x;
    float a0 = 0.0f, a1 = 0.0f;
    for (int p = t; p < CXP; p += 256) {
        a0 += colmax[p];
        a1 += colmax[CXP + p];
    }
    __shared__ float r0[256], r1[256];
    r0[t] = a0;
    r1[t] = a1;
    __syncthreads();
    for (int st = 128; st > 0; st >>= 1) {
        if (t < st) { r0[t] += r0[t + st]; r1[t] += r1[t + st]; }
        __syncthreads();
    }
    if (t == 0) {
        const float l0 = -logf(r0[0] * (1.0f / (float)CXP));
        const float l1 = -logf(r1[0] * (1.0f / (float)CXP));
        out[0] = 0.5f * (l0 + l1);
    }
}

// ---------------------------------------------------------------------------
// Launch wrapper
// ---------------------------------------------------------------------------
extern "C" void kernel_launch(void* const* d_in, const int* in_sizes, int n_in,
                              void* d_out, int out_size, void* d_ws, size_t ws_size,
                              hipStream_t stream) {
    const float* fT = (const float*)d_in[0]; // featureT [2,256,64,64]
    const float* fI = (const float*)d_in[1]; // featureI [2,256,64,64]
    float* out = (float*)d_out;

    // Workspace layout (floats): ~16.9 MB total
    float* ws     = (float*)d_ws;
    float* meanT  = ws;                                  // 256
    float* QN     = meanT + CXC;                         // 2*4096*256 (normalized fI)
    float* TN     = QN + (size_t)CXN * CXP * CXC;        // 2*4096*256 (normalized fT)
    float* rowmin = TN + (size_t)CXN * CXP * CXC;        // 8192
    float* rowsum = rowmin + NPOS;                       // 8192
    float* colmax = rowsum + NPOS;                       // 8192

    cx_mean_kernel<<<CXC, 256, 0, stream>>>(fT, meanT);
    cx_normalize_kernel<<<NPOS, 256, 0, stream>>>(fI, meanT, QN);
    cx_normalize_kernel<<<NPOS, 256, 0, stream>>>(fT, meanT, TN);
    cx_init_kernel<<<(NPOS + 255) / 256, 256, 0, stream>>>(rowmin, rowsum, colmax);

    // 64x64 tiles: (4096/64)^2 = 4096 blocks per batch image, x2 images
    const int gemm_blocks = CXN * 64 * 64; // 8192
    cx_gemm_pass<1><<<gemm_blocks, 512, 0, stream>>>(QN, TN, rowmin, rowsum, colmax);
    cx_gemm_pass<2><<<gemm_blocks, 512, 0, stream>>>(QN, TN, rowmin, rowsum, colmax);
    cx_gemm_pass<3><<<gemm_blocks, 512, 0, stream>>>(QN, TN, rowmin, rowsum, colmax);

    cx_final_kernel<<<1, 256, 0, stream>>>(colmax, out);
}